// LSEPLoss_33157147525792
// MI455X (gfx1250) — compile-verified
//
#include <hip/hip_runtime.h>
#include <math.h>

// LSEP loss: out = log1p( sum_r (sum_{t=0} e^{x}) * (sum_{t>0} e^{-x}) ) / B
// 384 MB streamed once -> HBM-bound (~16.5 us floor @ 23.3 TB/s).
// Strategy: NT b128 loads, in-lane 24-wide row reduction, wave32 shuffle
// reduce, fixed-size block partials (deterministic), tiny finalize kernel.

typedef float v4f __attribute__((ext_vector_type(4)));
typedef int   v4i __attribute__((ext_vector_type(4)));

#define LSEP_C        24
#define LSEP_NBLOCKS  1024
#define LSEP_NTHREADS 256

__global__ __launch_bounds__(LSEP_NTHREADS)
void lsep_partial_kernel(const float* __restrict__ in,
                         const int*   __restrict__ tgt,
                         float*       __restrict__ partial,
                         int nrows)
{
    const int tid    = blockIdx.x * blockDim.x + threadIdx.x;
    const int stride = gridDim.x * blockDim.x;

    float acc = 0.0f;

    for (int r = tid; r < nrows; r += stride) {
        // Row r: 24 floats = 96 bytes, 16-byte aligned -> 6 x b128 loads each.
        const v4f* xp = (const v4f*)(in  + (size_t)r * LSEP_C);
        const v4i* tp = (const v4i*)(tgt + (size_t)r * LSEP_C);

        float s_pos = 0.0f;  // sum over t>0 of exp(-x)
        float s_neg = 0.0f;  // sum over t==0 of exp(x)

        #pragma unroll
        for (int q = 0; q < LSEP_C / 4; ++q) {
            // Non-temporal: 384 MB stream with zero reuse; do not pollute L2.
            v4f x = __builtin_nontemporal_load(xp + q);
            v4i t = __builtin_nontemporal_load(tp + q);
            #pragma unroll
            for (int e = 0; e < 4; ++e) {
                const float xv = x[e];
                const int   tv = t[e];
                // Exactly one exp per element regardless of sign selection.
                const float ev = __expf(tv > 0 ? -xv : xv);
                if (tv > 0)       s_pos += ev;
                else if (tv == 0) s_neg += ev;
            }
        }
        acc += s_pos * s_neg;
    }

    // ---- wave32 reduction (warpSize == 32 on gfx1250) ----
    #pragma unroll
    for (int off = 16; off > 0; off >>= 1)
        acc += __shfl_down(acc, off, 32);

    __shared__ float smem[LSEP_NTHREADS / 32];
    const int lane = threadIdx.x & 31;
    const int wave = threadIdx.x >> 5;
    if (lane == 0) smem[wave] = acc;
    __syncthreads();

    if (wave == 0) {
        float v = (lane < (LSEP_NTHREADS / 32)) ? smem[lane] : 0.0f;
        #pragma unroll
        for (int off = (LSEP_NTHREADS / 64); off > 0; off >>= 1)
            v += __shfl_down(v, off, 32);
        if (lane == 0) partial[blockIdx.x] = v;  // fully overwritten every call
    }
}

__global__ __launch_bounds__(1024)
void lsep_final_kernel(const float* __restrict__ partial,
                       float*       __restrict__ out,
                       float inv_batch)
{
    // One block of 1024 threads reduces the 1024 block partials.
    __shared__ float smem[32];
    const int lane = threadIdx.x & 31;
    const int wave = threadIdx.x >> 5;

    float v = partial[threadIdx.x];
    #pragma unroll
    for (int off = 16; off > 0; off >>= 1)
        v += __shfl_down(v, off, 32);
    if (lane == 0) smem[wave] = v;
    __syncthreads();

    if (wave == 0) {
        float w = smem[lane];  // exactly 32 waves -> all lanes valid
        #pragma unroll
        for (int off = 16; off > 0; off >>= 1)
            w += __shfl_down(w, off, 32);
        if (lane == 0)
            out[0] = log1pf(w) * inv_batch;
    }
}

extern "C" void kernel_launch(void* const* d_in, const int* in_sizes, int n_in,
                              void* d_out, int out_size, void* d_ws, size_t ws_size,
                              hipStream_t stream)
{
    const float* in  = (const float*)d_in[0];
    const int*   tgt = (const int*)  d_in[1];
    float*       out = (float*)d_out;
    float*       ws  = (float*)d_ws;        // LSEP_NBLOCKS floats of scratch

    const int nrows = in_sizes[0] / LSEP_C; // 2,000,000

    lsep_partial_kernel<<<LSEP_NBLOCKS, LSEP_NTHREADS, 0, stream>>>(
        in, tgt, ws, nrows);

    lsep_final_kernel<<<1, 1024, 0, stream>>>(
        ws, out, 1.0f / (float)nrows);
}